// OptimizedSpectralPredictor_23218593202872
// MI455X (gfx1250) — compile-verified
//
#include <hip/hip_runtime.h>

// Problem constants (from reference)
constexpr int ZB = 224, YB = 512, XB = 512;
constexpr int PW = 15;               // spectral window
constexpr int WD = PW + 4;           // 19 weights per band
constexpr float MAXV =  32767.0f;
constexpr float MINV = -32768.0f;

// Tiling: 32x8 threads, each thread owns 2 adjacent rows -> 32x16 pixel tile
constexpr int TX = 32, TY = 8, RPT = 2;
constexpr int TILE_H = TY * RPT;     // 16
constexpr int HW = TX + 1;           // 33 (halo width)
constexpr int HH = TILE_H + 1;       // 17 (halo height)
constexpr int ELEMS = HW * HH;       // 561 halo elements per band tile
constexpr int NBUF = 3;              // triple buffer, prefetch depth 2

typedef float v2f __attribute__((ext_vector_type(2)));
// Constant-address-space pointer: uniform loads lower to s_load_* (SGPR broadcast)
typedef __attribute__((address_space(4))) const float* cwptr;

__global__ void zero_word_kernel(float* z) {
    if (threadIdx.x == 0 && blockIdx.x == 0) *z = 0.0f;
}

__global__ __launch_bounds__(256)
void spec_pred_kernel(const float* __restrict__ img,
                      const float* __restrict__ wts,
                      const float* __restrict__ zbuf,   // 1 float, == 0.0f
                      float* __restrict__ preds,
                      float* __restrict__ resids)
{
    __shared__ float tiles[NBUF][ELEMS];

    const int tx  = threadIdx.x, ty = threadIdx.y;
    const int t   = ty * TX + tx;                // 0..255
    const int x0  = blockIdx.x * TX;
    const int y0  = blockIdx.y * TILE_H;
    const int gx  = x0 + tx;
    const int ty2 = ty * RPT;
    const int gy0 = y0 + ty2;

    const size_t NPIX = (size_t)YB * XB;

    // LDS byte offsets of the three tile buffers (low 32 bits of flat addr = LDS offset)
    unsigned ldsb[NBUF];
#pragma unroll
    for (int b = 0; b < NBUF; ++b)
        ldsb[b] = (unsigned)(unsigned long long)(&tiles[b][0]);

    // 3 halo slots per thread (last clamped, never EXEC-masked -> every wave issues
    // exactly 3 async instructions per tile => exact ASYNCcnt accounting).
    // Out-of-image halo slots read a device zero word instead: the LDS halo then
    // holds exact zeros where the reference zero-pads -> no per-band predicates.
    const int s0 = t;
    const int s1 = t + 256;                                    // < 561 always
    const int s2 = (t + 512 < ELEMS) ? (t + 512) : (ELEMS - 1);

    const float* gp[3];
    size_t       st[3];
    {
        const int ss[3] = { s0, s1, s2 };
#pragma unroll
        for (int k = 0; k < 3; ++k) {
            const int r = ss[k] / HW, c = ss[k] % HW;
            const int gyy = y0 - 1 + r;
            const int gxx = x0 - 1 + c;
            const bool valid = (gyy >= 0) && (gxx >= 0);
            gp[k] = valid ? (img + (size_t)gyy * XB + gxx) : zbuf;
            st[k] = valid ? NPIX : 0;       // zero slot never advances
        }
    }
    const unsigned so0 = (unsigned)s0 * 4u;
    const unsigned so1 = (unsigned)s1 * 4u;
    const unsigned so2 = (unsigned)s2 * 4u;

    // CDNA5 async global->LDS tile prefetch (tracked by ASYNCcnt); running pointers
    auto prefetch = [&](unsigned base) {
        asm volatile("global_load_async_to_lds_b32 %0, %1, off"
                     :: "v"(base + so0), "v"(gp[0]) : "memory");
        asm volatile("global_load_async_to_lds_b32 %0, %1, off"
                     :: "v"(base + so1), "v"(gp[1]) : "memory");
        asm volatile("global_load_async_to_lds_b32 %0, %1, off"
                     :: "v"(base + so2), "v"(gp[2]) : "memory");
        gp[0] += st[0]; gp[1] += st[1]; gp[2] += st[2];
    };

    // Weights via constant address space -> s_load_* into SGPRs, KMcnt-tracked.
    cwptr cw = (cwptr)(unsigned long long)wts;

    // Spectral ring (bands z-15..z-1), phase-indexed: slot i holds band == i (mod 15).
    v2f rv[PW];
#pragma unroll
    for (int j = 0; j < PW; ++j) { rv[j].x = 0.0f; rv[j].y = 0.0f; }

    unsigned o = (unsigned)((size_t)gy0 * XB + gx);   // running output offset

    prefetch(ldsb[0]);
    prefetch(ldsb[1]);

// One band step. PH folds to a constant after unrolling (ring & buffer phases).
// Order: wait(own tile loads) -> barrier(all waves' loads) -> prefetch band+2
// (overwrites the buffer everyone consumed one step ago, provably before this
// barrier) -> compute.
#define STEP(PH, DO_PF, IS_LAST) do {                                          \
    if (IS_LAST) { asm volatile("s_wait_asynccnt 0x0" ::: "memory"); }         \
    else         { asm volatile("s_wait_asynccnt 0x3" ::: "memory"); }         \
    __syncthreads();                                                           \
    if (DO_PF) prefetch(ldsb[((PH) + 2) % NBUF]);                              \
    const float* T_ = &tiles[(PH) % NBUF][0];                                  \
    const float cur0 = T_[(ty2 + 1) * HW + tx + 1];                            \
    const float cur1 = T_[(ty2 + 2) * HW + tx + 1];                            \
    const float wl0  = T_[(ty2 + 1) * HW + tx    ];                            \
    const float wl1  = T_[(ty2 + 2) * HW + tx    ];                            \
    const float nl0  = T_[ ty2      * HW + tx + 1];                            \
    const float nwl0 = T_[ ty2      * HW + tx    ];                            \
    v2f cv, nv, wv, nwv;                                                       \
    cv.x  = cur0;  cv.y  = cur1;                                               \
    nv.x  = nl0;   nv.y  = cur0;    /* row1 north = row0 cur  */               \
    wv.x  = wl0;   wv.y  = wl1;                                                \
    nwv.x = nwl0;  nwv.y = wl0;     /* row1 nw    = row0 west */               \
    const float w33 = cw[3] * (1.0f / 3.0f);                                   \
    v2f p = nv * cw[0] + wv * cw[1] + nwv * cw[2] + (nv + wv + nwv) * w33;     \
    _Pragma("unroll")                                                          \
    for (int j = 0; j < PW; ++j) p += rv[((PH) + j) % PW] * cw[4 + j];         \
    p.x = fminf(fmaxf(p.x, MINV), MAXV);                                       \
    p.y = fminf(fmaxf(p.y, MINV), MAXV);                                       \
    v2f res = cv - p;                                                          \
    __builtin_nontemporal_store(p.x,   preds  + o);                            \
    __builtin_nontemporal_store(p.y,   preds  + o + XB);                       \
    __builtin_nontemporal_store(res.x, resids + o);                            \
    __builtin_nontemporal_store(res.y, resids + o + XB);                       \
    rv[(PH) % PW] = cv;             /* band z replaces band z-15 */            \
    o  += (unsigned)NPIX;                                                      \
    cw += WD;                                                                  \
} while (0)

    // 224 = 14*15 + 14; 15 | (zo*15) and 15 | 210, 3 | 210 -> phases fold.
    for (int zo = 0; zo < 14; ++zo) {
#pragma unroll
        for (int zi = 0; zi < PW; ++zi)
            STEP(zi, true, false);
    }
#pragma unroll
    for (int zi = 0; zi < 14; ++zi)
        STEP(zi, (zi < 12), (zi == 13));

#undef STEP
}

extern "C" void kernel_launch(void* const* d_in, const int* in_sizes, int n_in,
                              void* d_out, int out_size, void* d_ws, size_t ws_size,
                              hipStream_t stream) {
    (void)in_sizes; (void)n_in; (void)ws_size; (void)out_size;
    const float* img = (const float*)d_in[0];   // image   (Z,Y,X) f32
    const float* wts = (const float*)d_in[1];   // weights (Z,19)  f32
    // d_in[2] = local_sums: carry-only in the reference, never affects the outputs.

    float* zbuf   = (float*)d_ws;               // 4 bytes of scratch -> zero word
    float* preds  = (float*)d_out;
    float* resids = preds + (size_t)ZB * YB * XB;

    zero_word_kernel<<<1, 32, 0, stream>>>(zbuf);

    dim3 grid(XB / TX, YB / TILE_H);   // 16 x 32 blocks
    dim3 block(TX, TY);                // 32 x 8 = 256 threads (8 waves, wave32)
    spec_pred_kernel<<<grid, block, 0, stream>>>(img, wts, zbuf, preds, resids);
}